// HFMSA_22136261443992
// MI455X (gfx1250) — compile-verified
//
#include <hip/hip_runtime.h>
#include <math.h>

// ---------------- problem constants ----------------
#define B_ 8
#define C_ 256
#define H_ 128
#define W_ 128
#define HW_ (H_ * W_)          // 16384
#define O1_ 16                 // C/RED

typedef __attribute__((ext_vector_type(2))) float v2f;
typedef __attribute__((ext_vector_type(8))) float v8f;
typedef __attribute__((ext_vector_type(4))) unsigned int v4u;
typedef __attribute__((ext_vector_type(4))) int v4i;
typedef __attribute__((ext_vector_type(8))) int v8i;

__device__ __forceinline__ unsigned br7(unsigned x) { return __brev(x) >> 25; }

// fp32 WMMA 16x16x4: D = A(16x4) * B(4x16) + C(16x16)
__device__ __forceinline__ v8f wmma_f32(v2f a, v2f b, v8f c) {
    return __builtin_amdgcn_wmma_f32_16x16x4_f32(false, a, false, b, (short)0, c,
                                                 false, false);
}

// ---- Tensor Data Mover: async 2D tile load (global -> LDS), ISA 08 D# packing ----
// Loads tile [tile_d1 rows x tile_d0 elems] of fp32 from a row-strided tensor
// (row stride = stride0 elems) starting at gaddr, into LDS at byte offset lds_off
// (row-major, tile_d0*4 bytes per row). Tracked by TENSORcnt.
__device__ __forceinline__ void tdm_load_2d(unsigned lds_off, const float* gaddr,
                                            unsigned tile_d0, unsigned tile_d1,
                                            unsigned tensor_d0, unsigned tensor_d1,
                                            unsigned long long stride0) {
    unsigned long long ga = (unsigned long long)(uintptr_t)gaddr;
    v4u g0;
    g0.x = 1u;                                            // count=1 (valid), no gather
    g0.y = lds_off;                                       // LDS byte address
    g0.z = (unsigned)(ga & 0xffffffffu);                  // global_addr[31:0]
    g0.w = (unsigned)((ga >> 32) & 0x1ffffffu) | (2u << 30);  // addr[56:32] | type=2
    v8i g1;
    g1[0] = (int)(2u << 16);                              // wg_mask=0, data_size=2 (4B)
    g1[1] = (int)((tensor_d0 & 0xffffu) << 16);           // barrier_addr=0 | dim0.lo16
    g1[2] = (int)((tensor_d0 >> 16) | ((tensor_d1 & 0xffffu) << 16));
    g1[3] = (int)((tensor_d1 >> 16) | (tile_d0 << 16));   // dim1.hi16 | tile_dim0
    g1[4] = (int)tile_d1;                                 // tile_dim1 (tile_dim2=0)
    g1[5] = (int)(unsigned)(stride0 & 0xffffffffull);     // dim0_stride[31:0]
    g1[6] = (int)(unsigned)((stride0 >> 32) & 0xffffull); // dim0_stride[47:32]
    g1[7] = 0;                                            // dim1_stride (unused, 2D)
    v4i z4 = {0, 0, 0, 0};
#if __clang_major__ >= 23
    v8i z8 = {0, 0, 0, 0, 0, 0, 0, 0};
    __builtin_amdgcn_tensor_load_to_lds(g0, g1, z4, z4, z8, 0);
#else
    __builtin_amdgcn_tensor_load_to_lds(g0, g1, z4, z4, 0);
#endif
}

// ============ K1: multi-scale depthwise dilated conv, ms = mean of 3 ============
__global__ void k_dwconv(const float* __restrict__ x, const float* __restrict__ w1,
                         const float* __restrict__ w2, const float* __restrict__ w3,
                         float* __restrict__ ms) {
    size_t idx = (size_t)blockIdx.x * blockDim.x + threadIdx.x;
    int wc = (int)(idx & 127);
    int h  = (int)((idx >> 7) & 127);
    int c  = (int)((idx >> 14) & 255);
    const float* xp = x + (idx & ~(size_t)(HW_ - 1));  // plane base
    const float* wts[3] = {w1 + c * 9, w2 + c * 9, w3 + c * 9};
    const int dil[3] = {1, 2, 4};
    float acc = 0.f;
#pragma unroll
    for (int p = 0; p < 3; ++p) {
        const int d = dil[p];
        const float* wt = wts[p];
#pragma unroll
        for (int ky = 0; ky < 3; ++ky) {
            int hh = h + (ky - 1) * d;
            if ((unsigned)hh >= 128u) continue;
#pragma unroll
            for (int kx = 0; kx < 3; ++kx) {
                int ww = wc + (kx - 1) * d;
                if ((unsigned)ww >= 128u) continue;
                acc += wt[ky * 3 + kx] * xp[(hh << 7) + ww];
            }
        }
    }
    ms[idx] = acc * (1.0f / 3.0f);
}

// ============ K2: per-plane 128x128 FFT2 -> |.|, fftshift over ALL axes ============
// One workgroup per (b,c) plane; radix-2 DIT in 128 KB dynamic LDS (CDNA5: 320KB/WGP).
__global__ void k_fft_mag(const float* __restrict__ ms, float* __restrict__ hf) {
    extern __shared__ float2 sm[];  // 16384 complex = 128 KB
    const int tid = threadIdx.x;    // 512 threads
    const int img = blockIdx.x;     // b*256 + c
    const int b = img >> 8, c = img & 255;
    const float* src = ms + (size_t)img * HW_;

    for (int i = tid; i < HW_; i += 512) {   // load, columns bit-reversed
        int row = i >> 7, col = i & 127;
        float2 v; v.x = src[i]; v.y = 0.f;
        sm[(row << 7) | br7((unsigned)col)] = v;
    }
    __syncthreads();

    for (int s = 0; s < 7; ++s) {            // row FFTs
        int half = 1 << s;
        for (int t = tid; t < 8192; t += 512) {
            int row = t >> 6, j = t & 63;
            int grp = j >> s, pos = j & (half - 1);
            int i0 = (row << 7) + (grp << (s + 1)) + pos;
            int i1 = i0 + half;
            float ang = -3.14159265358979f * (float)pos / (float)half;
            float sn, cs; __sincosf(ang, &sn, &cs);
            float2 u = sm[i0], v = sm[i1];
            float tr = cs * v.x - sn * v.y;
            float ti = cs * v.y + sn * v.x;
            sm[i1] = make_float2(u.x - tr, u.y - ti);
            sm[i0] = make_float2(u.x + tr, u.y + ti);
        }
        __syncthreads();
    }

    for (int i = tid; i < HW_; i += 512) {   // bit-reverse rows
        int row = i >> 7, col = i & 127;
        int rr = (int)br7((unsigned)row);
        if (row < rr) {
            float2 tmp = sm[(row << 7) + col];
            sm[(row << 7) + col] = sm[(rr << 7) + col];
            sm[(rr << 7) + col] = tmp;
        }
    }
    __syncthreads();

    for (int s = 0; s < 7; ++s) {            // column FFTs
        int half = 1 << s;
        for (int t = tid; t < 8192; t += 512) {
            int col = t & 127, j = t >> 7;
            int grp = j >> s, pos = j & (half - 1);
            int i0 = (((grp << (s + 1)) + pos) << 7) + col;
            int i1 = i0 + (half << 7);
            float ang = -3.14159265358979f * (float)pos / (float)half;
            float sn, cs; __sincosf(ang, &sn, &cs);
            float2 u = sm[i0], v = sm[i1];
            float tr = cs * v.x - sn * v.y;
            float ti = cs * v.y + sn * v.x;
            sm[i1] = make_float2(u.x - tr, u.y - ti);
            sm[i0] = make_float2(u.x + tr, u.y + ti);
        }
        __syncthreads();
    }

    // magnitude + fftshift on all 4 axes folded into the write index
    const int b2 = (b + 4) & 7, c2 = (c + 128) & 255;
    float* dst = hf + (size_t)(b2 * 256 + c2) * HW_;
    for (int i = tid; i < HW_; i += 512) {
        int row = i >> 7, col = i & 127;
        float2 v = sm[i];
        dst[(((row + 64) & 127) << 7) | ((col + 64) & 127)] = sqrtf(v.x * v.x + v.y * v.y);
    }
}

// ============ K3: h1[b,o,p] = relu( sum_c ft_w1[o,c] * hf[b,c,p] ) ============
// TDM loads the 256x128 fp32 B panel (128 KB) into LDS; WMMA fp32 from LDS.
// grid: B*128 blocks x 256 threads (8 waves); dyn LDS = 16KB (A) + 128KB (B).
__global__ void k_ft1(const float* __restrict__ hf, const float* __restrict__ ft_w1,
                      float* __restrict__ h1) {
    extern __shared__ float sm3[];
    float* Ws = sm3;          // 4096 floats: ft_w1 [16][256]
    float* Bl = sm3 + 4096;   // 32768 floats: panel [256 c][128 pix]

    const int b = blockIdx.x >> 7, grp = blockIdx.x & 127;
    for (int i = threadIdx.x; i < 16 * 256; i += 256) Ws[i] = ft_w1[i];
    if (threadIdx.x < 32) {   // wave 0 issues the async tensor DMA
        tdm_load_2d((unsigned)(uintptr_t)Bl,
                    hf + (size_t)b * C_ * HW_ + grp * 128,
                    /*tile_d0=*/128, /*tile_d1=*/C_,
                    /*tensor_d0=*/HW_, /*tensor_d1=*/C_, /*stride0=*/HW_);
        __builtin_amdgcn_s_wait_tensorcnt(0);
    }
    __syncthreads();

    const int wave = threadIdx.x >> 5, lane = threadIdx.x & 31;
    const int m = lane & 15, n = lane & 15;
    const int koff = (lane >> 4) << 1;       // lanes 16-31 hold K+2,K+3
    const int pixl = wave * 16 + n;          // local pixel 0..127

    v8f acc = {};
#pragma unroll 4
    for (int k = 0; k < 256; k += 4) {
        v2f a, bv;
        a.x = Ws[m * 256 + k + koff];
        a.y = Ws[m * 256 + k + koff + 1];
        bv.x = Bl[(k + koff) * 128 + pixl];
        bv.y = Bl[(k + koff + 1) * 128 + pixl];
        acc = wmma_f32(a, bv, acc);
    }
    const int oBase = (lane >> 4) << 3;
    float* outp = h1 + (size_t)b * O1_ * HW_ + grp * 128 + wave * 16 + n;
#pragma unroll
    for (int i = 0; i < 8; ++i)
        outp[(size_t)(oBase + i) * HW_] = fmaxf(acc[i], 0.f);
}

// ============ K4: ft_attn = sigmoid(ft_w2 @ h1); ef = ms * ft_attn (WMMA fp32) ============
__global__ void k_ft2_ef(const float* __restrict__ h1, const float* __restrict__ ft_w2,
                         const float* __restrict__ ms, float* __restrict__ ef) {
    const int blk = blockIdx.x;
    const int b = blk >> 11, mtile = (blk >> 7) & 15, ngrp = blk & 127;
    const int wave = threadIdx.x >> 5, lane = threadIdx.x & 31;
    const int pix0 = (ngrp * 8 + wave) * 16;
    const int m = lane & 15, n = lane & 15;
    const int koff = (lane >> 4) << 1;

    const int cBase = mtile * 16 + ((lane >> 4) << 3);
    const float* msp = ms + (size_t)b * C_ * HW_ + pix0 + n;
    float* efp = ef + (size_t)b * C_ * HW_ + pix0 + n;
#pragma unroll
    for (int i = 0; i < 8; ++i)
        __builtin_prefetch(msp + (size_t)(cBase + i) * HW_, 0, 0);  // global_prefetch_b8

    const float* Bp = h1 + (size_t)b * O1_ * HW_ + pix0 + n;
    v8f acc = {};
#pragma unroll
    for (int k = 0; k < 16; k += 4) {
        v2f a, bv;
        a.x = ft_w2[(mtile * 16 + m) * 16 + k + koff];
        a.y = ft_w2[(mtile * 16 + m) * 16 + k + koff + 1];
        bv.x = Bp[(size_t)(k + koff) * HW_];
        bv.y = Bp[(size_t)(k + koff + 1) * HW_];
        acc = wmma_f32(a, bv, acc);
    }
#pragma unroll
    for (int i = 0; i < 8; ++i) {
        size_t off = (size_t)(cBase + i) * HW_;
        float attn = 1.f / (1.f + expf(-acc[i]));
        efp[off] = msp[off] * attn;
    }
}

// ============ K5: per-(b,c) avg/max/min pooling over HxW ============
__global__ void k_pool(const float* __restrict__ ef, float* __restrict__ pooled) {
    __shared__ float rs[256], rmx[256], rmn[256];
    const int bc = blockIdx.x;  // 0..2047
    const float* p = ef + (size_t)bc * HW_;
    float s = 0.f, mx = -3.4e38f, mn = 3.4e38f;
    for (int i = threadIdx.x; i < HW_; i += 256) {
        float v = p[i];
        s += v; mx = fmaxf(mx, v); mn = fminf(mn, v);
    }
    rs[threadIdx.x] = s; rmx[threadIdx.x] = mx; rmn[threadIdx.x] = mn;
    __syncthreads();
    for (int st = 128; st > 0; st >>= 1) {
        if (threadIdx.x < st) {
            rs[threadIdx.x] += rs[threadIdx.x + st];
            rmx[threadIdx.x] = fmaxf(rmx[threadIdx.x], rmx[threadIdx.x + st]);
            rmn[threadIdx.x] = fminf(rmn[threadIdx.x], rmn[threadIdx.x + st]);
        }
        __syncthreads();
    }
    if (threadIdx.x == 0) {
        int b = bc >> 8, c = bc & 255;
        pooled[b * 768 + c]       = rs[0] * (1.0f / HW_);
        pooled[b * 768 + 256 + c] = rmx[0];
        pooled[b * 768 + 512 + c] = rmn[0];
    }
}

// ============ K6: ca = sigmoid(relu(pooled @ w1^T) @ w2^T) ============
__global__ void k_ca(const float* __restrict__ pooled, const float* __restrict__ w1,
                     const float* __restrict__ w2, float* __restrict__ ca) {
    __shared__ float hid[B_ * 16];
    const int t = threadIdx.x;
    if (t < B_ * 16) {
        int b = t >> 4, j = t & 15;
        float s = 0.f;
        for (int i = 0; i < 768; ++i) s += pooled[b * 768 + i] * w1[j * 768 + i];
        hid[t] = fmaxf(s, 0.f);
    }
    __syncthreads();
    for (int idx = t; idx < B_ * C_; idx += 256) {
        int b = idx >> 8, c = idx & 255;
        float s = 0.f;
#pragma unroll
        for (int j = 0; j < 16; ++j) s += hid[b * 16 + j] * w2[c * 16 + j];
        ca[idx] = 1.f / (1.f + expf(-s));
    }
}

// ============ K7: out = fuse_w @ (ef*ca) + fuse_b + x ============
// One block owns a (b, 128-pixel) panel: ef panel TDM-loaded into LDS ONCE,
// then all 16 M-tiles (full C_out=256) computed against it with WMMA fp32.
// dyn LDS = 16KB (A' tile) + 128KB (B panel) = 144KB (fits 2 WGs in 320KB WGP).
__global__ void k_fuse(const float* __restrict__ ef, const float* __restrict__ fuse_w,
                       const float* __restrict__ fuse_b, const float* __restrict__ ca,
                       const float* __restrict__ x, float* __restrict__ out) {
    extern __shared__ float sm7[];
    float* As = sm7;          // 4096 floats: A'[16][256] for current mtile
    float* Bl = sm7 + 4096;   // 32768 floats: ef panel [256 c][128 pix]

    const int b = blockIdx.x >> 7, ngrp = blockIdx.x & 127;
    if (threadIdx.x < 32) {   // wave 0: async tensor DMA of the whole K-panel
        tdm_load_2d((unsigned)(uintptr_t)Bl,
                    ef + (size_t)b * C_ * HW_ + ngrp * 128,
                    /*tile_d0=*/128, /*tile_d1=*/C_,
                    /*tensor_d0=*/HW_, /*tensor_d1=*/C_, /*stride0=*/HW_);
        __builtin_amdgcn_s_wait_tensorcnt(0);
    }

    const int wave = threadIdx.x >> 5, lane = threadIdx.x & 31;
    const int m = lane & 15, n = lane & 15;
    const int koff = (lane >> 4) << 1;
    const int pixl = wave * 16 + n;
    const size_t pbase = (size_t)b * C_ * HW_ + ngrp * 128 + wave * 16 + n;

    for (int mt = 0; mt < 16; ++mt) {
        __syncthreads();  // panel ready (mt==0) / previous round done with As
        // A'[o,c] = fuse_w[o,c] * ca[b,c] : fold channel attention into weights
        for (int i = threadIdx.x; i < 16 * 256; i += 256) {
            int mm = i >> 8, cc = i & 255;
            As[i] = fuse_w[(mt * 16 + mm) * 256 + cc] * ca[b * 256 + cc];
        }
        __syncthreads();

        v8f acc = {};
#pragma unroll 4
        for (int k = 0; k < 256; k += 4) {
            v2f a, bv;
            a.x = As[m * 256 + k + koff];
            a.y = As[m * 256 + k + koff + 1];
            bv.x = Bl[(k + koff) * 128 + pixl];
            bv.y = Bl[(k + koff + 1) * 128 + pixl];
            acc = wmma_f32(a, bv, acc);
        }
        const int oBase = mt * 16 + ((lane >> 4) << 3);
#pragma unroll
        for (int i = 0; i < 8; ++i) {
            int o = oBase + i;
            size_t off = pbase + (size_t)o * HW_;
            out[off] = acc[i] + fuse_b[o] + x[off];
        }
    }
}

// ================================= launcher =================================
extern "C" void kernel_launch(void* const* d_in, const int* in_sizes, int n_in,
                              void* d_out, int out_size, void* d_ws, size_t ws_size,
                              hipStream_t stream) {
    const float* x      = (const float*)d_in[0];
    const float* w_d1   = (const float*)d_in[1];
    const float* w_d2   = (const float*)d_in[2];
    const float* w_d3   = (const float*)d_in[3];
    const float* ft_w1  = (const float*)d_in[4];
    const float* ft_w2  = (const float*)d_in[5];
    const float* ca_w1  = (const float*)d_in[6];
    const float* ca_w2  = (const float*)d_in[7];
    const float* fuse_w = (const float*)d_in[8];
    const float* fuse_b = (const float*)d_in[9];
    float* out = (float*)d_out;

    // workspace (floats): ms | hf (reused as ef after K3) | h1 | pooled | ca
    const size_t TENS = (size_t)B_ * C_ * HW_;  // 33,554,432 floats (134 MB)
    float* ms     = (float*)d_ws;
    float* hf     = ms + TENS;
    float* ef     = hf;  // hf dead after K3; reuse for ef
    float* h1     = hf + TENS;
    float* pooled = h1 + (size_t)B_ * O1_ * HW_;
    float* ca     = pooled + B_ * 3 * C_;

    const size_t lds_mm = (16 * 256 + 256 * 128) * sizeof(float);  // 147456 B

    k_dwconv<<<(int)(TENS / 256), 256, 0, stream>>>(x, w_d1, w_d2, w_d3, ms);
    k_fft_mag<<<B_ * C_, 512, HW_ * sizeof(float2), stream>>>(ms, hf);
    k_ft1<<<B_ * 128, 256, lds_mm, stream>>>(hf, ft_w1, h1);
    k_ft2_ef<<<B_ * 16 * 128, 256, 0, stream>>>(h1, ft_w2, ms, ef);
    k_pool<<<B_ * C_, 256, 0, stream>>>(ef, pooled);
    k_ca<<<1, 256, 0, stream>>>(pooled, ca_w1, ca_w2, ca);
    k_fuse<<<B_ * 128, 256, lds_mm, stream>>>(ef, fuse_w, fuse_b, ca, x, out);
}